// Net_time_23398981828939
// MI455X (gfx1250) — compile-verified
//
#include <hip/hip_runtime.h>
#include <hip/hip_bf16.h>

// ---------------------------------------------------------------------------
// ST-GCN style block for MI455X (gfx1250), fp32 throughout using the native
// fp32 matrix pipe: V_WMMA_F32_16X16X4_F32.
// ---------------------------------------------------------------------------

typedef __attribute__((ext_vector_type(2))) float v2f;
typedef __attribute__((ext_vector_type(8))) float v8f;

__device__ __forceinline__ v8f wmma4(v2f a, v2f b, v8f c) {
    // 8 args: (neg_a, A, neg_b, B, c_mod, C, reuse_a, reuse_b)
    return __builtin_amdgcn_wmma_f32_16x16x4_f32(false, a, false, b,
                                                 (short)0, c, false, false);
}

#define Bn  64
#define Tn  300
#define Vn  25
#define Cn  64
#define NE  (64LL * 300LL * 25LL * 64LL)   // 30,720,000 elements
#define NSAMP 480000.0f                     // B*T*V samples per channel

// ---------------------------------------------------------------------------
// K0: normalized adjacencies  An = D^-1/2 (fill_diag(A*E,1)) D^-1/2 , bsum
// ---------------------------------------------------------------------------
__global__ void prep_kernel(const float* __restrict__ A, const float* __restrict__ E,
                            const float* __restrict__ b1, const float* __restrict__ b2,
                            const float* __restrict__ b3,
                            float* __restrict__ An, float* __restrict__ bsum) {
    __shared__ float dinv_s[3][Vn];
    int tid = threadIdx.x;
    if (tid < 3 * Vn) {
        int br = tid / Vn, v = tid % Vn;
        float d = 0.f;
        for (int u = 0; u < Vn; ++u) {
            float ad = (u == v) ? 1.0f : A[br * 625 + v * 25 + u] * E[br * 625 + v * 25 + u];
            d += ad;
        }
        dinv_s[br][v] = (d > 0.f) ? rsqrtf(d) : 0.f;
    }
    if (tid < Cn) bsum[tid] = b1[tid] + b2[tid] + b3[tid];
    __syncthreads();
    for (int idx = tid; idx < 3 * 625; idx += blockDim.x) {
        int br = idx / 625, rem = idx % 625;
        int v = rem / 25, u = rem % 25;
        float ad = (u == v) ? 1.0f : A[idx] * E[idx];
        An[idx] = dinv_s[br][v] * ad * dinv_s[br][u];
    }
}

// ---------------------------------------------------------------------------
// K1: spatial block, one workgroup (4 waves) per (b,t).
//     z[b,t] = sum_i An_i @ (X @ W_i) + bsum   (V padded 25 -> 32)
// ---------------------------------------------------------------------------
__global__ __launch_bounds__(128) void spatial_kernel(
        const float* __restrict__ x,
        const float* __restrict__ W1, const float* __restrict__ W2,
        const float* __restrict__ W3,
        const float* __restrict__ An, const float* __restrict__ bsum,
        float* __restrict__ z) {
    __shared__ float Xs[32 * 68];   // X tile (rows >=25 zero), padded stride
    __shared__ float Ps[32 * 68];   // P = X @ W_i
    __shared__ float Ans[32 * 36];  // adjacency tile, padded stride

    const int t = blockIdx.x, b = blockIdx.y;
    const int tid = threadIdx.x;
    const long long bt = (long long)b * Tn + t;
    const float* xptr = x + bt * (Vn * Cn);

    for (int idx = tid; idx < 32 * 64; idx += 128) {
        int r = idx >> 6, cc = idx & 63;
        Xs[r * 68 + cc] = (r < Vn) ? xptr[r * 64 + cc] : 0.f;
    }

    const int lane = tid & 31, wave = tid >> 5;
    const int lane16 = lane & 15, khalf = lane >> 4;
    const int c = wave * 16 + lane16;           // output channel column

    v8f acc0 = {}, acc1 = {};
    const float* Ws[3] = {W1, W2, W3};

    for (int br = 0; br < 3; ++br) {
        __syncthreads();   // prior-branch LDS consumers done
        for (int idx = tid; idx < 32 * 32; idx += 128) {
            int r = idx >> 5, cc = idx & 31;
            Ans[r * 36 + cc] = (r < Vn && cc < Vn) ? An[br * 625 + r * 25 + cc] : 0.f;
        }
        const float* W = Ws[br];

        // P = X @ W   (M=32, N=64 split over 4 waves, K=64)
        v8f p0 = {}, p1 = {};
        for (int k0 = 0; k0 < 64; k0 += 4) {
            int kk = k0 + 2 * khalf;
            v2f bf = { W[kk * 64 + c], W[(kk + 1) * 64 + c] };
            v2f a0 = { Xs[lane16 * 68 + kk], Xs[lane16 * 68 + kk + 1] };
            v2f a1 = { Xs[(16 + lane16) * 68 + kk], Xs[(16 + lane16) * 68 + kk + 1] };
            p0 = wmma4(a0, bf, p0);
            p1 = wmma4(a1, bf, p1);
        }
        for (int r = 0; r < 8; ++r) {
            Ps[(r + 8 * khalf) * 68 + c]      = p0[r];
            Ps[(16 + r + 8 * khalf) * 68 + c] = p1[r];
        }
        __syncthreads();

        // acc += An_br @ P   (K=32)
        for (int k0 = 0; k0 < 32; k0 += 4) {
            int kk = k0 + 2 * khalf;
            v2f bf = { Ps[kk * 68 + c], Ps[(kk + 1) * 68 + c] };
            v2f a0 = { Ans[lane16 * 36 + kk], Ans[lane16 * 36 + kk + 1] };
            v2f a1 = { Ans[(16 + lane16) * 36 + kk], Ans[(16 + lane16) * 36 + kk + 1] };
            acc0 = wmma4(a0, bf, acc0);
            acc1 = wmma4(a1, bf, acc1);
        }
    }

    const float bc = bsum[c];
    float* zptr = z + bt * (Vn * Cn);
    for (int r = 0; r < 8; ++r) {
        int v0 = r + 8 * khalf;
        if (v0 < Vn) zptr[v0 * 64 + c] = acc0[r] + bc;
        int v1 = 16 + r + 8 * khalf;
        if (v1 < Vn) zptr[v1 * 64 + c] = acc1[r] + bc;
    }
}

// ---------------------------------------------------------------------------
// K2/K6: per-channel sum / sumsq partials (fixed 256x256 grid, deterministic)
// stride (65536) is a multiple of 64, so each thread owns one channel.
// ---------------------------------------------------------------------------
__global__ __launch_bounds__(256) void stats_kernel(const float* __restrict__ z,
                                                    float* __restrict__ partial,
                                                    long long n) {
    long long stride = (long long)gridDim.x * blockDim.x;
    long long i = (long long)blockIdx.x * blockDim.x + threadIdx.x;
    float s = 0.f, ss = 0.f;
    for (; i < n; i += stride) { float v = z[i]; s += v; ss += v * v; }
    __shared__ float sh[2][256];
    sh[0][threadIdx.x] = s; sh[1][threadIdx.x] = ss;
    __syncthreads();
    if (threadIdx.x < 64) {
        int cix = threadIdx.x;
        float ts  = sh[0][cix] + sh[0][cix + 64] + sh[0][cix + 128] + sh[0][cix + 192];
        float tss = sh[1][cix] + sh[1][cix + 64] + sh[1][cix + 128] + sh[1][cix + 192];
        partial[(blockIdx.x * 2 + 0) * 64 + cix] = ts;
        partial[(blockIdx.x * 2 + 1) * 64 + cix] = tss;
    }
}

// K3/K7: fold BN into per-channel scale/shift
__global__ void finalize_kernel(const float* __restrict__ partial,
                                const float* __restrict__ gamma,
                                const float* __restrict__ beta,
                                float* __restrict__ scale, float* __restrict__ shift,
                                float invN) {
    int cix = threadIdx.x;   // 64 threads
    float s = 0.f, ss = 0.f;
    for (int bkt = 0; bkt < 256; ++bkt) {
        s  += partial[(bkt * 2 + 0) * 64 + cix];
        ss += partial[(bkt * 2 + 1) * 64 + cix];
    }
    float mean = s * invN;
    float var  = ss * invN - mean * mean;
    float sc   = gamma[cix] * rsqrtf(var + 1e-5f);
    scale[cix] = sc;
    shift[cix] = beta[cix] - mean * sc;
}

// ---------------------------------------------------------------------------
// K4: y = relu(bn(z)) @ Wt   — tiled WMMA GEMM, 64 rows x 64 cols per block,
// BN+ReLU fused into the LDS staging load.
// ---------------------------------------------------------------------------
__global__ __launch_bounds__(128) void tgemm_kernel(
        const float* __restrict__ z, const float* __restrict__ Wt,
        const float* __restrict__ scale, const float* __restrict__ shift,
        float* __restrict__ y) {
    __shared__ float As[64 * 68];
    const int tid = threadIdx.x;
    const long long row0 = (long long)blockIdx.x * 64;

    for (int idx = tid; idx < 64 * 64; idx += 128) {
        int r = idx >> 6, cc = idx & 63;
        float h = z[(row0 + r) * 64 + cc] * scale[cc] + shift[cc];
        As[r * 68 + cc] = fmaxf(h, 0.f);
    }
    __syncthreads();

    const int lane = tid & 31, wave = tid >> 5;
    const int lane16 = lane & 15, khalf = lane >> 4;
    const int c = wave * 16 + lane16;

    v8f acc0 = {}, acc1 = {}, acc2 = {}, acc3 = {};
    for (int k0 = 0; k0 < 64; k0 += 4) {
        int kk = k0 + 2 * khalf;
        v2f bf = { Wt[kk * 64 + c], Wt[(kk + 1) * 64 + c] };
        v2f a;
        a = (v2f){ As[(     lane16) * 68 + kk], As[(     lane16) * 68 + kk + 1] }; acc0 = wmma4(a, bf, acc0);
        a = (v2f){ As[(16 + lane16) * 68 + kk], As[(16 + lane16) * 68 + kk + 1] }; acc1 = wmma4(a, bf, acc1);
        a = (v2f){ As[(32 + lane16) * 68 + kk], As[(32 + lane16) * 68 + kk + 1] }; acc2 = wmma4(a, bf, acc2);
        a = (v2f){ As[(48 + lane16) * 68 + kk], As[(48 + lane16) * 68 + kk + 1] }; acc3 = wmma4(a, bf, acc3);
    }
    for (int r = 0; r < 8; ++r) {
        int rr = r + 8 * khalf;
        y[(row0      + rr) * 64 + c] = acc0[r];
        y[(row0 + 16 + rr) * 64 + c] = acc1[r];
        y[(row0 + 32 + rr) * 64 + c] = acc2[r];
        y[(row0 + 48 + rr) * 64 + c] = acc3[r];
    }
}

// ---------------------------------------------------------------------------
// K5: temporal banded sum: u[b,t,v,c] = bt[c] + sum_{s=t-4..t+4 clamp} y[b,s,v,c]
// y (123 MB) is L2-resident (192 MB L2) so the 9x reread is cheap.
// ---------------------------------------------------------------------------
__global__ __launch_bounds__(256) void band_kernel(const float* __restrict__ y,
                                                   const float* __restrict__ btv,
                                                   float* __restrict__ u) {
    long long idx = (long long)blockIdx.x * blockDim.x + threadIdx.x;
    if (idx >= NE) return;
    int cc = (int)(idx & 63);
    long long row = idx >> 6;                 // (b*T + t)*V + v
    int v = (int)(row % Vn);
    long long bT = row / Vn;
    int t = (int)(bT % Tn);
    long long b = bT / Tn;
    int s0 = t - 4 < 0 ? 0 : t - 4;
    int s1 = t + 4 > Tn - 1 ? Tn - 1 : t + 4;
    const float* base = y + (b * Tn * Vn + v) * 64 + cc;
    float s = btv[cc];
    for (int ss = s0; ss <= s1; ++ss) s += base[(long long)ss * (Vn * 64)];
    u[idx] = s;
}

// K8: final BN + ReLU
__global__ __launch_bounds__(256) void act_kernel(const float* __restrict__ u,
                                                  const float* __restrict__ scale,
                                                  const float* __restrict__ shift,
                                                  float* __restrict__ out, long long n) {
    long long idx = (long long)blockIdx.x * blockDim.x + threadIdx.x;
    if (idx >= n) return;
    int cc = (int)(idx & 63);
    out[idx] = fmaxf(u[idx] * scale[cc] + shift[cc], 0.f);
}

// ---------------------------------------------------------------------------
extern "C" void kernel_launch(void* const* d_in, const int* in_sizes, int n_in,
                              void* d_out, int out_size, void* d_ws, size_t ws_size,
                              hipStream_t stream) {
    const float* x   = (const float*)d_in[0];
    const float* A   = (const float*)d_in[1];
    const float* E   = (const float*)d_in[2];
    const float* W1  = (const float*)d_in[3];
    const float* b1  = (const float*)d_in[4];
    const float* W2  = (const float*)d_in[5];
    const float* b2  = (const float*)d_in[6];
    const float* W3  = (const float*)d_in[7];
    const float* b3  = (const float*)d_in[8];
    const float* Wt  = (const float*)d_in[9];
    const float* bt  = (const float*)d_in[10];
    const float* gam = (const float*)d_in[11];
    const float* bet = (const float*)d_in[12];
    float* out = (float*)d_out;

    float* ws   = (float*)d_ws;
    float* zbuf = ws;                 // z, later reused for u  (30,720,000 f)
    float* An   = ws + NE;            // 1875 (padded to 2048)
    float* bsum = An + 2048;          // 64
    float* part = bsum + 64;          // 256*2*64 = 32768
    float* sc1  = part + 32768;       // 64
    float* sh1  = sc1 + 64;
    float* sc2  = sh1 + 64;
    float* sh2  = sc2 + 64;

    const float invN = 1.0f / NSAMP;
    const int nElemBlocks = (int)((NE + 255) / 256);

    prep_kernel<<<1, 128, 0, stream>>>(A, E, b1, b2, b3, An, bsum);
    spatial_kernel<<<dim3(Tn, Bn), 128, 0, stream>>>(x, W1, W2, W3, An, bsum, zbuf);
    stats_kernel<<<256, 256, 0, stream>>>(zbuf, part, NE);
    finalize_kernel<<<1, 64, 0, stream>>>(part, gam, bet, sc1, sh1, invN);
    tgemm_kernel<<<(int)(NE / 64 / 64), 128, 0, stream>>>(zbuf, Wt, sc1, sh1, out); // y -> d_out
    band_kernel<<<nElemBlocks, 256, 0, stream>>>(out, bt, zbuf);                    // u -> zbuf
    stats_kernel<<<256, 256, 0, stream>>>(zbuf, part, NE);
    finalize_kernel<<<1, 64, 0, stream>>>(part, gam, bet, sc2, sh2, invN);
    act_kernel<<<nElemBlocks, 256, 0, stream>>>(zbuf, sc2, sh2, out, NE);
}